// BoundaryLoss_50689204028104
// MI455X (gfx1250) — compile-verified
//
#include <hip/hip_runtime.h>
#include <float.h>
#include <math.h>

// Problem constants (from reference setup_inputs)
#define B_    8
#define C_    3
#define H_    384
#define W_    384
#define HW_   (H_ * W_)
#define NINST (B_ * C_)     // 24 (b,c) instances
#define NMAPS (NINST * 2)   // 48 distance maps (bg + fg per instance)
#define INF_  1.0e6f

#define GRID_C 1024
#define BLK_C  256

typedef __attribute__((ext_vector_type(2))) float v2f;
typedef __attribute__((ext_vector_type(8))) float v8f;

// ---------------------------------------------------------------------------
// Exact f32 wave-sum via V_WMMA_F32_16X16X4_F32.
// A = ones(16x4), B holds one partial per lane (b.x = acc, b.y = 0).
// D[m,n] = sum_k B[k,n]  (independent of m) -> every lane's c[0] holds a
// column sum; summing c[0] over all 32 lanes double-counts (rows 0 and 8 of D
// both appear), so halve at the end. All math stays in fp32 (exact reduction
// tree, deterministic).
// ---------------------------------------------------------------------------
__device__ __forceinline__ float wave_sum32(float x) {
  v2f a; a.x = 1.0f; a.y = 1.0f;
  v2f b; b.x = x;    b.y = 0.0f;
  v8f c = {};
  c = __builtin_amdgcn_wmma_f32_16x16x4_f32(false, a, false, b, (short)0, c,
                                            false, false);
  float s = c[0];
  s += __shfl_xor(s, 16);
  s += __shfl_xor(s, 8);
  s += __shfl_xor(s, 4);
  s += __shfl_xor(s, 2);
  s += __shfl_xor(s, 1);
  return 0.5f * s;
}

__device__ __forceinline__ float wave_max32(float v) {
  v = fmaxf(v, __shfl_xor(v, 16));
  v = fmaxf(v, __shfl_xor(v, 8));
  v = fmaxf(v, __shfl_xor(v, 4));
  v = fmaxf(v, __shfl_xor(v, 2));
  v = fmaxf(v, __shfl_xor(v, 1));
  return v;
}

// ---------------------------------------------------------------------------
// Kernel 0: zero the max / any buffers (workspace is poisoned with 0xAA).
// ---------------------------------------------------------------------------
__global__ void k_init(unsigned* maxbuf, int* anyflag) {
  int t = threadIdx.x;
  if (t < NMAPS) maxbuf[t] = 0u;
  if (t < NINST) anyflag[t] = 0;
}

// ---------------------------------------------------------------------------
// Kernel 1: per-column 1D distance scans (axis 0), for BOTH mask=(tgt==c)
// (bg map, t=0) and its complement (fg map, t=1). One thread per column;
// column mask bits packed into 12 regs so targets is read once.
// Matches reference exactly: d = bit ? 0 : d+1 with d0 = 1e6 (fp32 exact).
// ---------------------------------------------------------------------------
__global__ void k_colscan(const int* __restrict__ targets,
                          float* __restrict__ g,
                          int* __restrict__ anyflag) {
  int inst = blockIdx.x / 3;       // (b,c)
  int jblk = blockIdx.x % 3;
  int j    = jblk * 128 + threadIdx.x;
  int b    = inst / C_;
  int c    = inst % C_;

  const int* tp = targets + (size_t)b * HW_ + j;

  unsigned mbits[12];
  unsigned anyb = 0;
  for (int w = 0; w < 12; ++w) {
    unsigned mw = 0;
    for (int r = 0; r < 32; ++r) {
      int i = w * 32 + r;
      mw |= (unsigned)(tp[i * W_] == c) << r;
    }
    mbits[w] = mw;
    anyb |= mw;
  }
  if (anyb) anyflag[inst] = 1;   // benign same-value race, deterministic

  for (int t = 0; t < 2; ++t) {
    float* gm = g + (size_t)(inst * 2 + t) * HW_ + j;
    float d = INF_;
    for (int i = 0; i < H_; ++i) {
      unsigned bitv = (mbits[i >> 5] >> (i & 31)) & 1u;
      if (t) bitv ^= 1u;
      d = bitv ? 0.0f : d + 1.0f;
      gm[(size_t)i * W_] = d;
    }
    d = INF_;
    for (int i = H_ - 1; i >= 0; --i) {
      unsigned bitv = (mbits[i >> 5] >> (i & 31)) & 1u;
      if (t) bitv ^= 1u;
      d = bitv ? 0.0f : d + 1.0f;
      float p = gm[(size_t)i * W_];
      gm[(size_t)i * W_] = fminf(p, d);
    }
  }
}

// ---------------------------------------------------------------------------
// Kernel 2: exact row pass  d[i,j] = sqrt(min_k g[i,k]^2 + (j-k)^2).
// One wave per row (12 outputs/lane), row cached in LDS as g^2, in-place
// overwrite of g with the final distance. fmaf(dk,dk,g2) reproduces the
// reference rounding ((j-k)^2 is an exact f32 integer). Per-map max via
// ordered-uint atomicMax (distances >= 0 -> uint order == float order).
// ---------------------------------------------------------------------------
__global__ void k_rowedt(float* __restrict__ g, unsigned* __restrict__ maxbuf) {
  __shared__ float srow[8][W_];
  int wid  = threadIdx.x >> 5;
  int lane = threadIdx.x & 31;
  int row  = blockIdx.x * 8 + wid;
  int m    = row / H_;
  int i    = row % H_;

  float* rp = g + (size_t)m * HW_ + (size_t)i * W_;

  #pragma unroll
  for (int u = 0; u < 12; ++u) {
    float v = rp[lane + 32 * u];
    srow[wid][lane + 32 * u] = v * v;
  }
  __syncthreads();

  float jf[12], d2[12];
  #pragma unroll
  for (int u = 0; u < 12; ++u) {
    jf[u] = (float)(lane + 32 * u);
    d2[u] = FLT_MAX;
  }

  const float* s = srow[wid];
  for (int k = 0; k < W_; ++k) {
    float g2 = s[k];            // broadcast LDS read
    float kf = (float)k;
    #pragma unroll
    for (int u = 0; u < 12; ++u) {
      float dk = jf[u] - kf;
      d2[u] = fminf(d2[u], fmaf(dk, dk, g2));
    }
  }

  float mx = 0.0f;
  #pragma unroll
  for (int u = 0; u < 12; ++u) {
    float d = sqrtf(d2[u]);
    rp[lane + 32 * u] = d;      // in-place: g buffer becomes dist buffer
    mx = fmaxf(mx, d);
  }
  mx = wave_max32(mx);
  if (lane == 0) atomicMax(&maxbuf[m], __float_as_uint(mx));
}

// ---------------------------------------------------------------------------
// Kernel 3: fused softmax * normalized SDF, grid-stride accumulate.
// Wave sums via f32 WMMA, fixed-tree block reduce -> per-block partial.
// ---------------------------------------------------------------------------
__global__ void k_loss(const float* __restrict__ logits,
                       const float* __restrict__ dist,
                       const unsigned* __restrict__ maxbuf,
                       const int* __restrict__ anyflag,
                       float* __restrict__ partials) {
  const int N = B_ * HW_;
  float acc = 0.0f;
  for (int idx = blockIdx.x * BLK_C + threadIdx.x; idx < N;
       idx += GRID_C * BLK_C) {
    int b  = idx / HW_;
    int hw = idx - b * HW_;
    const float* lp = logits + (size_t)b * C_ * HW_ + hw;
    float l0 = lp[0], l1 = lp[(size_t)HW_], l2 = lp[(size_t)2 * HW_];
    float mx = fmaxf(l0, fmaxf(l1, l2));
    float e0 = expf(l0 - mx), e1 = expf(l1 - mx), e2 = expf(l2 - mx);
    float inv = 1.0f / (e0 + e1 + e2);
    float p0 = e0 * inv, p1 = e1 * inv, p2 = e2 * inv;
    float pc[3] = {p0, p1, p2};
    #pragma unroll
    for (int c = 0; c < C_; ++c) {
      int inst = b * C_ + c;
      if (anyflag[inst]) {
        float mbg = fmaxf(__uint_as_float(maxbuf[2 * inst]), 1e-12f);
        float mfg = fmaxf(__uint_as_float(maxbuf[2 * inst + 1]), 1e-12f);
        float dbg = dist[(size_t)(2 * inst) * HW_ + hw];
        float dfg = dist[(size_t)(2 * inst + 1) * HW_ + hw];
        float sdf = dbg / mbg - dfg / mfg;
        acc += pc[c] * sdf;
      }
    }
  }

  float ws = wave_sum32(acc);   // EXEC is full here (loop has converged)
  __shared__ float lpart[BLK_C / 32];
  if ((threadIdx.x & 31) == 0) lpart[threadIdx.x >> 5] = ws;
  __syncthreads();
  if (threadIdx.x < 32) {
    float v = (threadIdx.x < BLK_C / 32) ? lpart[threadIdx.x] : 0.0f;
    v += __shfl_xor(v, 16);
    v += __shfl_xor(v, 8);
    v += __shfl_xor(v, 4);
    v += __shfl_xor(v, 2);
    v += __shfl_xor(v, 1);
    if (threadIdx.x == 0) partials[blockIdx.x] = v;
  }
}

// ---------------------------------------------------------------------------
// Kernel 4: final deterministic reduction of GRID_C partials -> mean.
// ---------------------------------------------------------------------------
__global__ void k_finish(const float* __restrict__ partials,
                         float* __restrict__ out) {
  float acc = 0.0f;
  for (int p = threadIdx.x; p < GRID_C; p += BLK_C) acc += partials[p];
  float ws = wave_sum32(acc);
  __shared__ float lpart[BLK_C / 32];
  if ((threadIdx.x & 31) == 0) lpart[threadIdx.x >> 5] = ws;
  __syncthreads();
  if (threadIdx.x < 32) {
    float v = (threadIdx.x < BLK_C / 32) ? lpart[threadIdx.x] : 0.0f;
    v += __shfl_xor(v, 16);
    v += __shfl_xor(v, 8);
    v += __shfl_xor(v, 4);
    v += __shfl_xor(v, 2);
    v += __shfl_xor(v, 1);
    if (threadIdx.x == 0)
      out[0] = v * (1.0f / (float)(B_ * C_ * HW_));
  }
}

// ---------------------------------------------------------------------------
extern "C" void kernel_launch(void* const* d_in, const int* in_sizes, int n_in,
                              void* d_out, int out_size, void* d_ws,
                              size_t ws_size, hipStream_t stream) {
  const float* logits  = (const float*)d_in[0];   // [8,3,384,384] f32
  const int*   targets = (const int*)d_in[1];     // [8,384,384] i32

  char*     ws       = (char*)d_ws;
  float*    g        = (float*)ws;                               // 48*HW f32 (g, then dist in-place)
  unsigned* maxbuf   = (unsigned*)(ws + (size_t)NMAPS * HW_ * 4);
  int*      anyflag  = (int*)((char*)maxbuf + NMAPS * 4);
  float*    partials = (float*)((char*)anyflag + NINST * 4);

  k_init   <<<1, 128, 0, stream>>>(maxbuf, anyflag);
  k_colscan<<<NINST * 3, 128, 0, stream>>>(targets, g, anyflag);
  k_rowedt <<<(NMAPS * H_) / 8, 256, 0, stream>>>(g, maxbuf);
  k_loss   <<<GRID_C, BLK_C, 0, stream>>>(logits, g, maxbuf, anyflag, partials);
  k_finish <<<1, BLK_C, 0, stream>>>(partials, (float*)d_out);
}